// GCN_3624952398755
// MI455X (gfx1250) — compile-verified
//
#include <hip/hip_runtime.h>
#include <math.h>

typedef __attribute__((ext_vector_type(2))) float v2f;
typedef __attribute__((ext_vector_type(8))) float v8f;

#define NNODES 100000
#define HID    128
#define TILE_M 16
#define ROWT   2                    // row tiles per transform block
#define ROWS   (ROWT * TILE_M)      // 32 rows/block; 100000 = 32 * 3125

static __device__ __forceinline__ float atomic_add_f32(float* p, float v) {
  return __hip_atomic_fetch_add(p, v, __ATOMIC_RELAXED, __HIP_MEMORY_SCOPE_AGENT);
}

// ---------------- degree kernels ----------------
__global__ __launch_bounds__(256) void deg_kernel(const int* __restrict__ dst,
                                                  float* __restrict__ deg, int E) {
  int e = blockIdx.x * 256 + threadIdx.x;
  if (e < E) atomic_add_f32(&deg[dst[e]], 1.0f);
}

__global__ __launch_bounds__(256) void invdeg_kernel(float* __restrict__ deg, int N) {
  int n = blockIdx.x * 256 + threadIdx.x;
  if (n < N) deg[n] = 1.0f / fmaxf(deg[n], 1.0f);
}

// ---------------- edge scatter-sum: agg[dst] += x[src] ----------------
__global__ __launch_bounds__(256) void scatter_kernel(const float* __restrict__ x,
                                                      const int* __restrict__ src,
                                                      const int* __restrict__ dst,
                                                      float* __restrict__ agg, int E) {
  int gid  = blockIdx.x * 256 + threadIdx.x;
  int e    = gid >> 5;
  int lane = gid & 31;
  if (e >= E) return;
  int s = src[e], d = dst[e];
  const float4 v = *(const float4*)(x + (size_t)s * HID + lane * 4);
  float* ap = agg + (size_t)d * HID + lane * 4;
  atomic_add_f32(ap + 0, v.x);
  atomic_add_f32(ap + 1, v.y);
  atomic_add_f32(ap + 2, v.z);
  atomic_add_f32(ap + 3, v.w);
}

// ---------------- SAGE layer ----------------
__global__ __launch_bounds__(256) void sage_transform(
    const float* __restrict__ agg, const float* __restrict__ xin,
    const float* __restrict__ Wl, const float* __restrict__ bl,
    const float* __restrict__ Wr, const float* __restrict__ invdeg,
    float* __restrict__ xout) {
  __shared__ float lds_a[ROWS * HID];
  __shared__ float lds_x[ROWS * HID];
  __shared__ float lds_o[ROWS * HID];
  __shared__ float lds_inv[ROWS];

  const int tid  = threadIdx.x;
  const int row0 = blockIdx.x * ROWS;

#pragma unroll
  for (int rt = 0; rt < ROWT; ++rt) {
    const int r = rt * TILE_M + (tid >> 4);
    const int c = (tid & 15) * 8;
    const float inv = invdeg[row0 + r];
    const float4* pa = (const float4*)(agg + (size_t)(row0 + r) * HID + c);
    float4 a0 = pa[0], a1 = pa[1];
    float4* la4 = (float4*)(lds_a + r * HID + c);
    la4[0] = make_float4(a0.x * inv, a0.y * inv, a0.z * inv, a0.w * inv);
    la4[1] = make_float4(a1.x * inv, a1.y * inv, a1.z * inv, a1.w * inv);
    const float4* px = (const float4*)(xin + (size_t)(row0 + r) * HID + c);
    float4* lx4 = (float4*)(lds_x + r * HID + c);
    lx4[0] = px[0];
    lx4[1] = px[1];
  }

  const int wave  = tid >> 5;
  const int lane  = tid & 31;
  const int nn    = lane & 15;
  const int khalf = (lane >> 4) * 2;
  const int h0    = wave * 16;

  v2f bL[32], bR[32];
  {
    const float* pwl = Wl + (size_t)(h0 + nn) * HID + khalf;
    const float* pwr = Wr + (size_t)(h0 + nn) * HID + khalf;
#pragma unroll
    for (int kk = 0; kk < 32; ++kk) {
      bL[kk] = *(const v2f*)(pwl + kk * 4);
      bR[kk] = *(const v2f*)(pwr + kk * 4);
    }
  }
  const float bias  = bl[h0 + nn];
  const int   rbase = (lane >> 4) * 8;
  __syncthreads();

#pragma unroll 1
  for (int rt = 0; rt < ROWT; ++rt) {
    v8f accL = {0.f, 0.f, 0.f, 0.f, 0.f, 0.f, 0.f, 0.f};
    v8f accR = {0.f, 0.f, 0.f, 0.f, 0.f, 0.f, 0.f, 0.f};
    const float* la = lds_a + (rt * TILE_M + nn) * HID + khalf;
    const float* lx = lds_x + (rt * TILE_M + nn) * HID + khalf;
#pragma unroll
    for (int kk = 0; kk < 32; ++kk) {
      v2f aA = *(const v2f*)(la + kk * 4);
      accL = __builtin_amdgcn_wmma_f32_16x16x4_f32(false, aA, false, bL[kk],
                                                   (short)0, accL, false, false);
      v2f aX = *(const v2f*)(lx + kk * 4);
      accR = __builtin_amdgcn_wmma_f32_16x16x4_f32(false, aX, false, bR[kk],
                                                   (short)0, accR, false, false);
    }
#pragma unroll
    for (int j = 0; j < 8; ++j)
      lds_o[(rt * TILE_M + rbase + j) * HID + h0 + nn] = accL[j] + accR[j] + bias;
  }
  __syncthreads();

#pragma unroll
  for (int rt = 0; rt < ROWT; ++rt) {
    const int r = rt * TILE_M + (tid >> 4);
    const int p = tid & 15;
    const float* po = lds_o + r * HID + p * 8;
    float s = 0.f;
#pragma unroll
    for (int q = 0; q < 8; ++q) s += po[q] * po[q];
    s += __shfl_xor(s, 1);
    s += __shfl_xor(s, 2);
    s += __shfl_xor(s, 4);
    s += __shfl_xor(s, 8);
    if (p == 0) lds_inv[r] = 1.0f / fmaxf(sqrtf(s), 1e-12f);
  }
  __syncthreads();

#pragma unroll
  for (int rt = 0; rt < ROWT; ++rt) {
    const int r = rt * TILE_M + (tid >> 4);
    const int c = (tid & 15) * 8;
    const float inv = lds_inv[r];
    const float* po = lds_o + r * HID + c;
    float4 o0, o1;
    o0.x = fmaxf(po[0] * inv, 0.f); o0.y = fmaxf(po[1] * inv, 0.f);
    o0.z = fmaxf(po[2] * inv, 0.f); o0.w = fmaxf(po[3] * inv, 0.f);
    o1.x = fmaxf(po[4] * inv, 0.f); o1.y = fmaxf(po[5] * inv, 0.f);
    o1.z = fmaxf(po[6] * inv, 0.f); o1.w = fmaxf(po[7] * inv, 0.f);
    float4* pw = (float4*)(xout + (size_t)(row0 + r) * HID + c);
    pw[0] = o0;
    pw[1] = o1;
  }
}

// ---------------- MLP head ----------------
__global__ __launch_bounds__(256) void mlp_head(
    const float* __restrict__ xin, const float* __restrict__ W1,
    const float* __restrict__ b1, const float* __restrict__ W2,
    const float* __restrict__ b2, float* __restrict__ out) {
  __shared__ float lds_x[TILE_M * HID];
  __shared__ float lds_h[TILE_M * HID];

  const int tid  = threadIdx.x;
  const int row0 = blockIdx.x * TILE_M;

  {
    const int r = tid >> 4;
    const int c = (tid & 15) * 8;
    const float4* px = (const float4*)(xin + (size_t)(row0 + r) * HID + c);
    float4* lx4 = (float4*)(lds_x + r * HID + c);
    lx4[0] = px[0];
    lx4[1] = px[1];
  }

  const int wave  = tid >> 5;
  const int lane  = tid & 31;
  const int nn    = lane & 15;
  const int khalf = (lane >> 4) * 2;
  const int h0    = wave * 16;

  v2f bW[32];
  {
    const float* pw = W1 + (size_t)(h0 + nn) * HID + khalf;
#pragma unroll
    for (int kk = 0; kk < 32; ++kk) bW[kk] = *(const v2f*)(pw + kk * 4);
  }
  const float bias = b1[h0 + nn];
  __syncthreads();

  v8f acc = {0.f, 0.f, 0.f, 0.f, 0.f, 0.f, 0.f, 0.f};
  const float* lx = lds_x + nn * HID + khalf;
#pragma unroll
  for (int kk = 0; kk < 32; ++kk) {
    v2f a = *(const v2f*)(lx + kk * 4);
    acc = __builtin_amdgcn_wmma_f32_16x16x4_f32(false, a, false, bW[kk],
                                                (short)0, acc, false, false);
  }
  const int rbase = (lane >> 4) * 8;
#pragma unroll
  for (int j = 0; j < 8; ++j)
    lds_h[(rbase + j) * HID + h0 + nn] = fmaxf(acc[j] + bias, 0.f);
  __syncthreads();

  {
    const int r = tid >> 4;
    const int p = tid & 15;
    const float* ph = lds_h + r * HID + p * 8;
    float s0 = 0.f, s1 = 0.f;
#pragma unroll
    for (int q = 0; q < 8; ++q) {
      float h = ph[q];
      s0 += h * W2[p * 8 + q];
      s1 += h * W2[HID + p * 8 + q];
    }
    s0 += __shfl_xor(s0, 1); s1 += __shfl_xor(s1, 1);
    s0 += __shfl_xor(s0, 2); s1 += __shfl_xor(s1, 2);
    s0 += __shfl_xor(s0, 4); s1 += __shfl_xor(s1, 4);
    s0 += __shfl_xor(s0, 8); s1 += __shfl_xor(s1, 8);
    if (p == 0) {
      out[(size_t)(row0 + r) * 2 + 0] = s0 + b2[0];
      out[(size_t)(row0 + r) * 2 + 1] = s1 + b2[1];
    }
  }
}

// ---------------- launcher ----------------
extern "C" void kernel_launch(void* const* d_in, const int* in_sizes, int n_in,
                              void* d_out, int out_size, void* d_ws, size_t ws_size,
                              hipStream_t stream) {
  const float* x  = (const float*)d_in[0];
  const int*   ei = (const int*)d_in[1];
  const float* Wl = (const float*)d_in[2];
  const float* bl = (const float*)d_in[3];
  const float* Wr = (const float*)d_in[4];
  const float* W1 = (const float*)d_in[5];
  const float* b1 = (const float*)d_in[6];
  const float* W2 = (const float*)d_in[7];
  const float* b2 = (const float*)d_in[8];
  float* out = (float*)d_out;

  const int N = NNODES;
  const int E = in_sizes[1] / 2;
  const int* src = ei;
  const int* dst = ei + E;

  float* inv_deg = (float*)d_ws;
  const size_t pad = 100096;
  float* bufA = inv_deg + pad;
  float* bufB = bufA + (size_t)N * HID;

  hipMemsetAsync(inv_deg, 0, (size_t)N * sizeof(float), stream);
  deg_kernel<<<(E + 255) / 256, 256, 0, stream>>>(dst, inv_deg, E);
  invdeg_kernel<<<(N + 255) / 256, 256, 0, stream>>>(inv_deg, N);

  const float* xs[3] = {x, bufA, bufB};
  float*       ag[3] = {bufA, bufB, bufA};
  const int scatterBlocks = (E + 7) / 8;
  const int xformBlocks   = N / ROWS;     // 3125
  const int headBlocks    = N / TILE_M;   // 6250

  for (int i = 0; i < 3; ++i) {
    float* agg = ag[i];
    const float* xc = xs[i];
    hipMemsetAsync(agg, 0, (size_t)N * HID * sizeof(float), stream);
    scatter_kernel<<<scatterBlocks, 256, 0, stream>>>(xc, src, dst, agg, E);
    sage_transform<<<xformBlocks, 256, 0, stream>>>(
        agg, xc, Wl + (size_t)i * HID * HID, bl + (size_t)i * HID,
        Wr + (size_t)i * HID * HID, inv_deg, agg);
  }
  mlp_head<<<headBlocks, 256, 0, stream>>>(bufA, W1, b1, W2, b2, out);
}